// L_SSIM_12489764897415
// MI455X (gfx1250) — compile-verified
//
#include <hip/hip_runtime.h>

// ---------------------------------------------------------------------------
// SSIM fusion loss on MI455X (gfx1250), fp32, WMMA-based separable Gaussian.
// D = A x B + C with V_WMMA_F32_16X16X4_F32 (wave32).
// b64/b128 LDS traffic, hoisted addressing, double-buffered H tile,
// compiler-fence-only intra-wave LDS ordering (no per-field hardware drains).
// ---------------------------------------------------------------------------

typedef __attribute__((ext_vector_type(2))) float v2f;
typedef __attribute__((ext_vector_type(4))) float v4f;
typedef __attribute__((ext_vector_type(8))) float v8f;

#define IMG_H 512
#define IMG_W 512
#define TS    16          // output tile
#define HALO  26          // 16 + 2*5
#define KPAD  28          // halo padded to multiple of 4 (WMMA K chunks)
#define HSTR  36          // ldsHT row stride (floats): conflict-free, 16B-multiple

// aligned ds_load_b64 of LDS pair (k0 + 4c, k0 + 4c + 1)
#define LD2(p, c) (*(const v2f*)((p) + 4 * (c)))

#define WMMA4(a, b, c) __builtin_amdgcn_wmma_f32_16x16x4_f32( \
    false, (a), false, (b), (short)0, (c), false, false)

// Compiler-only memory fence: same-wave DS ops are processed in order by the
// LDS pipe (single-wave workgroup), so we only need to stop LLVM reordering.
#define WAVE_LDS_FENCE() asm volatile("" ::: "memory")

__device__ __forceinline__ float gauss_unnorm(int t) {
  float x = (float)(t - 5);
  return expf(-x * x * (1.0f / 4.5f));   // sigma = 1.5 -> 2*sigma^2 = 4.5
}

__global__ __launch_bounds__(32)
void ssim_tile_kernel(const float* __restrict__ A,
                      const float* __restrict__ B,
                      const float* __restrict__ F,
                      float* __restrict__ partial) {
  __shared__ float ldsA[HALO][KPAD];
  __shared__ float ldsB[HALO][KPAD];
  __shared__ float ldsF[HALO][KPAD];
  __shared__ float ldsHT[2][16][HSTR];   // transposed H buffer, double-buffered

  const int lane = threadIdx.x;          // 0..31 (wave32)
  const int tx = blockIdx.x, ty = blockIdx.y, p = blockIdx.z;
  const long plane = (long)p * (IMG_H * IMG_W);
  const float* __restrict__ pa = A + plane;
  const float* __restrict__ pb = B + plane;
  const float* __restrict__ pf = F + plane;

  // ---- Gaussian weights (normalized, matches reference window) ----
  float wsum = 0.f;
  #pragma unroll
  for (int t = 0; t < 11; ++t) wsum += gauss_unnorm(t);
  const float winv = 1.0f / wsum;

  // ---- constant banded-matrix fragments Wf[c]: W[k][idx] = w[k-idx].
  // A-fragment (lane->M, vgpr->K) and B-fragment (lane->N, vgpr->K) layouts
  // coincide, so one fragment set serves both conv passes.
  const int nidx = lane & 15;            // M (vertical A) / N (horizontal B)
  const int koff = (lane >> 4) << 1;     // lanes 16-31 hold K+2, K+3
  v2f Wf[7];
  #pragma unroll
  for (int c = 0; c < 7; ++c) {
    int k0 = c * 4 + koff;
    int d0 = k0 - nidx, d1 = k0 + 1 - nidx;
    Wf[c].x = (d0 >= 0 && d0 <= 10) ? gauss_unnorm(d0) * winv : 0.f;
    Wf[c].y = (d1 >= 0 && d1 <= 10) ? gauss_unnorm(d1) * winv : 0.f;
  }

  // ---- stage 26x28 halos of A, B, F (zero pad outside image AND cols 26/27,
  // so fragment loads need no column clamp and stay b64-contiguous) ----
  const int gr0 = ty * TS - 5, gc0 = tx * TS - 5;
  for (int t = lane; t < HALO * KPAD; t += 32) {
    int r = t / KPAD, cc = t - r * KPAD;
    int gr = gr0 + r, gc = gc0 + cc;
    bool ok = (cc < HALO) && (gr >= 0) && (gr < IMG_H) && (gc >= 0) && (gc < IMG_W);
    long off = (long)gr * IMG_W + gc;
    ldsA[r][cc] = ok ? pa[off] : 0.f;
    ldsB[r][cc] = ok ? pb[off] : 0.f;
    ldsF[r][cc] = ok ? pf[off] : 0.f;
  }
  __syncthreads();   // cross-lane visibility of staged halos

  // Hoisted per-lane row bases (row clamp only depends on the row-group).
  const int rr1 = (16 + nidx > HALO - 1) ? HALO - 1 : 16 + nidx;  // zero-weighted pad rows
  const float* bA0 = &ldsA[nidx][koff];
  const float* bA1 = &ldsA[rr1][koff];
  const float* bB0 = &ldsB[nidx][koff];
  const float* bB1 = &ldsB[rr1][koff];
  const float* bF0 = &ldsF[nidx][koff];
  const float* bF1 = &ldsF[rr1][koff];
  const int hi8 = (lane >> 4) << 3;      // D-layout: vgpr v -> row v + (lane>=16?8:0)

  // ---- 8 blurred fields: A, B, F, A^2, B^2, F^2, A*F, B*F ----
  v8f res[8];
  #pragma unroll
  for (int f = 0; f < 8; ++f) {
    const int buf = f & 1;
    // Horizontal pass: both 16-row groups with independent accumulators.
    v8f acc0 = {0.f, 0.f, 0.f, 0.f, 0.f, 0.f, 0.f, 0.f};
    v8f acc1 = {0.f, 0.f, 0.f, 0.f, 0.f, 0.f, 0.f, 0.f};
    #pragma unroll
    for (int c = 0; c < 7; ++c) {
      v2f a0, a1;
      switch (f) {
        case 0:  a0 = LD2(bA0, c);               a1 = LD2(bA1, c);               break;
        case 1:  a0 = LD2(bB0, c);               a1 = LD2(bB1, c);               break;
        case 2:  a0 = LD2(bF0, c);               a1 = LD2(bF1, c);               break;
        case 3:  { v2f t0 = LD2(bA0, c), t1 = LD2(bA1, c); a0 = t0 * t0; a1 = t1 * t1; } break;
        case 4:  { v2f t0 = LD2(bB0, c), t1 = LD2(bB1, c); a0 = t0 * t0; a1 = t1 * t1; } break;
        case 5:  { v2f t0 = LD2(bF0, c), t1 = LD2(bF1, c); a0 = t0 * t0; a1 = t1 * t1; } break;
        case 6:  a0 = LD2(bA0, c) * LD2(bF0, c); a1 = LD2(bA1, c) * LD2(bF1, c); break;
        default: a0 = LD2(bB0, c) * LD2(bF0, c); a1 = LD2(bB1, c) * LD2(bF1, c); break;
      }
      acc0 = WMMA4(a0, Wf[c], acc0);
      acc1 = WMMA4(a1, Wf[c], acc1);
    }
    // Transposed scatter: 8 contiguous floats per row-group -> 2x ds_store_b128.
    {
      float* hp = &ldsHT[buf][nidx][hi8];
      *(v4f*)(hp +  0) = acc0.lo;
      *(v4f*)(hp +  4) = acc0.hi;
      *(v4f*)(hp + 16) = acc1.lo;   // row-group 1 lives at k = 16..31
      *(v4f*)(hp + 20) = acc1.hi;
    }
    // hstore(f) -> vload(f): same buffer, per-lane overlapping ranges; the
    // in-order LDS pipe keeps RAW correct, we only fence the compiler.
    WAVE_LDS_FENCE();

    // Vertical pass: D = Wband(16x4) x H(4x16); even/odd accumulators for ILP.
    {
      const float* vb = &ldsHT[buf][nidx][koff];
      v8f ve = {0.f, 0.f, 0.f, 0.f, 0.f, 0.f, 0.f, 0.f};
      v8f vo = {0.f, 0.f, 0.f, 0.f, 0.f, 0.f, 0.f, 0.f};
      #pragma unroll
      for (int c = 0; c < 7; ++c) {
        v2f b = LD2(vb, c);
        if (c & 1) vo = WMMA4(Wf[c], b, vo);
        else       ve = WMMA4(Wf[c], b, ve);
      }
      res[f] = ve + vo;
    }
    WAVE_LDS_FENCE();  // vload(f) before hstore(f+2) reuses this buffer
  }

  // ---- pointwise SSIM for both pairs, 8 elements per lane ----
  const float c1 = 1e-4f, c2 = 9e-4f;    // 0.01^2, 0.03^2
  float accs = 0.f;
  #pragma unroll
  for (int j = 0; j < 8; ++j) {
    float muA = res[0][j], muB = res[1][j], muF = res[2][j];
    float eA2 = res[3][j], eB2 = res[4][j], eF2 = res[5][j];
    float eAF = res[6][j], eBF = res[7][j];
    float sA  = eA2 - muA * muA;
    float sB  = eB2 - muB * muB;
    float sF  = eF2 - muF * muF;
    float sAF = eAF - muA * muF;
    float sBF = eBF - muB * muF;
    float nAF = (2.f * muA * muF + c1) * (2.f * sAF + c2);
    float dAF = (muA * muA + muF * muF + c1) * (sA + sF + c2);
    float nBF = (2.f * muB * muF + c1) * (2.f * sBF + c2);
    float dBF = (muB * muB + muF * muF + c1) * (sB + sF + c2);
    accs += 0.5f * (nAF / dAF + nBF / dBF);
  }

  // ---- wave reduction (wave32) + per-block partial ----
  #pragma unroll
  for (int m = 16; m > 0; m >>= 1) accs += __shfl_xor(accs, m, 32);
  if (lane == 0)
    partial[((long)p * gridDim.y + ty) * gridDim.x + tx] = accs;
}

// Deterministic fixed-order final reduction: 32 waves, b128 loads, LDS tree.
__global__ __launch_bounds__(1024)
void ssim_reduce_kernel(const float* __restrict__ partial, int n4,
                        float* __restrict__ out, float inv_count) {
  __shared__ float sm[1024];
  const v4f* __restrict__ p4 = (const v4f*)partial;
  float a = 0.f;
  for (int i = threadIdx.x; i < n4; i += 1024) {
    v4f v = p4[i];
    a += (v.x + v.y) + (v.z + v.w);
  }
  sm[threadIdx.x] = a;
  __syncthreads();
  for (int st = 512; st > 0; st >>= 1) {
    if ((int)threadIdx.x < st) sm[threadIdx.x] += sm[threadIdx.x + st];
    __syncthreads();
  }
  if (threadIdx.x == 0) out[0] = sm[0] * inv_count;
}

extern "C" void kernel_launch(void* const* d_in, const int* in_sizes, int n_in,
                              void* d_out, int out_size, void* d_ws, size_t ws_size,
                              hipStream_t stream) {
  (void)n_in; (void)out_size; (void)ws_size;
  const float* A = (const float*)d_in[0];
  const float* B = (const float*)d_in[1];
  const float* F = (const float*)d_in[2];
  float* out     = (float*)d_out;
  float* partial = (float*)d_ws;        // needs planes*32*32*4 B = 192 KB scratch

  const int total  = in_sizes[0];                 // N*C*H*W
  const int planes = total / (IMG_H * IMG_W);     // N*C = 48
  dim3 grid(IMG_W / TS, IMG_H / TS, planes);      // 32 x 32 x 48 tiles

  ssim_tile_kernel<<<grid, 32, 0, stream>>>(A, B, F, partial);

  const int n_part = (IMG_W / TS) * (IMG_H / TS) * planes;  // multiple of 4
  ssim_reduce_kernel<<<1, 1024, 0, stream>>>(partial, n_part / 4, out,
                                             1.0f / (float)total);
}